// house_pred_55396488184383
// MI455X (gfx1250) — compile-verified
//
#include <hip/hip_runtime.h>

// ---------------- problem constants (from reference) ----------------
#define TOTAL 1000
#define INV   5
#define NUMF  8
#define EMB   100
#define BATCH 16384

#define K1 (NUMF * EMB)   // 800  (layer-1 K)
#define N1 100            // layer-1 out
#define N2 50             // layer-2 out
#define N3 25             // layer-3 out

// Padded / fragment-packed geometry
#define N1_PAD 112        // 7 WMMA N-tiles
#define N2_PAD 64         // 4 tiles
#define N3_PAD 32         // 2 tiles
#define K1_STEPS 200      // 800 / 4
#define K2_STEPS 25       // 100 / 4
#define K3_STEPS 13       // 52 / 4 (K padded 50 -> 52, zero-filled)

// LDS row strides (floats): even, with (stride/2) odd -> 16 row-parallel
// accesses hit 16 distinct banks; 8-byte alignment preserved.
#define H0_STRIDE 802
#define H1_STRIDE 114
#define H2_STRIDE 66
#define H3_STRIDE 34

// Scratch (d_ws) layout in floats
#define S_ELEMS    (TOTAL * EMB)                     // 100000
#define W1F_ELEMS  (K1_STEPS * 2 * N1_PAD * 2)       // 89600
#define W2F_ELEMS  (K2_STEPS * 2 * N2_PAD * 2)       // 6400
#define W3F_ELEMS  (K3_STEPS * 2 * N3_PAD * 2)       // 1664
#define S_OFF   0
#define W1F_OFF (S_OFF + S_ELEMS)
#define W2F_OFF (W1F_OFF + W1F_ELEMS)
#define W3F_OFF (W2F_OFF + W2F_ELEMS)
#define B1P_OFF (W3F_OFF + W3F_ELEMS)                // 112 floats, zero-padded
#define B2P_OFF (B1P_OFF + N1_PAD)                   // 64 floats
#define B3P_OFF (B2P_OFF + N2_PAD)                   // 32 floats
// total ~791.5 KB of d_ws, all L2-resident

typedef __attribute__((ext_vector_type(2))) float v2f;
typedef __attribute__((ext_vector_type(8))) float v8f;

// --------------------------------------------------------------------
// Kernel 1: collapse the one-hot box filter into a summed-embedding
// table  S[t][e] = sum_{j=0..10} w_emb[t + 995 + j][e].
// --------------------------------------------------------------------
__global__ void build_boxsum(const float* __restrict__ w_emb,
                             float* __restrict__ S) {
  int i = blockIdx.x * blockDim.x + threadIdx.x;
  if (i >= TOTAL * EMB) return;
  int t = i / EMB;
  int e = i - t * EMB;
  int k0 = t + TOTAL - INV;          // = t + 995; window never clips
  float acc = 0.f;
#pragma unroll
  for (int j = 0; j <= 2 * INV; ++j)
    acc += w_emb[(k0 + j) * EMB + e];
  S[i] = acc;
}

// --------------------------------------------------------------------
// Kernel 2: pre-swizzle a weight matrix [Kvalid x Nvalid] into exact
// wave32 B-fragment order, zero-padded:
//   wf[((g*2 + h)*Npad + n)*2 + j] = w[(4g + 2h + j)*Nvalid + n]
// so lane L of tile t loads its (b.x, b.y) pair as one contiguous,
// coalesced, unconditional global_load_b64. Padding baked in here.
// --------------------------------------------------------------------
__global__ void pack_bfrag(const float* __restrict__ w, float* __restrict__ wf,
                           int Kvalid, int Nvalid, int Npad, int Ksteps) {
  int i = blockIdx.x * blockDim.x + threadIdx.x;
  int total = Ksteps * 2 * Npad * 2;
  if (i >= total) return;
  int j  = i & 1;
  int n  = (i >> 1) % Npad;
  int gh = (i >> 1) / Npad;          // g*2 + h
  int h  = gh & 1;
  int g  = gh >> 1;
  int k  = g * 4 + 2 * h + j;
  wf[i] = (k < Kvalid && n < Nvalid) ? w[k * Nvalid + n] : 0.f;
}

// --------------------------------------------------------------------
// Kernel 3: zero-pad a bias vector so the GEMM epilogues need no guards:
// pad columns have accumulator 0 and bias 0 -> relu(0+0) = 0.
// --------------------------------------------------------------------
__global__ void pack_bias(const float* __restrict__ b, float* __restrict__ bp,
                          int Nvalid, int Npad) {
  int i = blockIdx.x * blockDim.x + threadIdx.x;
  if (i >= Npad) return;
  bp[i] = (i < Nvalid) ? b[i] : 0.f;
}

// --------------------------------------------------------------------
// Kernel 4: fused gather + 4-layer MLP, one workgroup (8 wave32) per 16
// batch rows. All GEMMs on V_WMMA_F32_16X16X4_F32; inner loop is
// ds_load_b64 (A) + global_load_b64 (B) + wmma, fully unconditional.
//
// Fragment layouts (ISA 7.12.2, wave32, 32-bit data):
//   A 16x4:  lanes 0-15: M=lane, v0=K0,v1=K1 ; lanes 16-31: M=lane-16, v0=K2,v1=K3
//   B 4x16:  lanes 0-15: N=lane, v0=K0,v1=K1 ; lanes 16-31: N=lane-16, v0=K2,v1=K3
//   C/D 16x16: VGPR r: lanes 0-15 -> (M=r, N=lane); lanes 16-31 -> (M=r+8, N=lane-16)
// --------------------------------------------------------------------
__launch_bounds__(256)
__global__ void mlp_fused(const int*   __restrict__ x,
                          const float* __restrict__ S,
                          const float* __restrict__ w1f, const float* __restrict__ b1p,
                          const float* __restrict__ w2f, const float* __restrict__ b2p,
                          const float* __restrict__ w3f, const float* __restrict__ b3p,
                          const float* __restrict__ w4,  const float* __restrict__ b4,
                          float* __restrict__ out) {
  __shared__ float h0[16 * H0_STRIDE];   // [16 x 800] activations
  __shared__ float h1[16 * H1_STRIDE];   // [16 x 100] (+zero pad to 112)
  __shared__ float h2[16 * H2_STRIDE];   // [16 x 50]  (+zero pad to 64)
  __shared__ float h3[16 * H3_STRIDE];   // [16 x 25]  (+zero pad to 32)

  const int tid  = threadIdx.x;
  const int wave = tid >> 5;
  const int lane = tid & 31;
  const int base = blockIdx.x * 16;      // first batch row of this tile

  const int half = lane >> 4;            // 0: lanes 0-15, 1: lanes 16-31
  const int mrow = lane & 15;            // A's M index == B's N index
  const int koff = half * 2;             // K sub-offset held by this half

  // ---- Phase 0: gather h0[m][f*EMB+e] = S[x[base+m][f]][e] -------------
  for (int i = tid; i < 16 * K1; i += 256) {
    int m = i / K1;
    int c = i - m * K1;
    int f = c / EMB;
    int e = c - f * EMB;
    int xv = x[(base + m) * NUMF + f];
    h0[m * H0_STRIDE + c] = S[xv * EMB + e];
  }
  __syncthreads();

  // ---- Layer 1: [16 x 800] @ w1[800 x 100], 7 N-tiles (waves 0-6) ------
  if (wave < 7) {
    const int ncol = wave * 16 + mrow;                 // < 112
    const float* ap = &h0[mrow * H0_STRIDE + koff];    // A: +4 floats/step
    const float* bp = &w1f[(half * N1_PAD + ncol) * 2];// B: +2*N1_PAD*2/step
    v8f c = {};
#pragma unroll 8
    for (int g = 0; g < K1_STEPS; ++g) {
      v2f a = *(const v2f*)(ap + g * 4);
      v2f b = *(const v2f*)(bp + g * (2 * N1_PAD * 2));
      c = __builtin_amdgcn_wmma_f32_16x16x4_f32(false, a, false, b,
                                                (short)0, c, false, false);
    }
    const float bias = b1p[ncol];        // zero in pad region
#pragma unroll
    for (int r = 0; r < 8; ++r) {
      int m = r + half * 8;
      h1[m * H1_STRIDE + ncol] = fmaxf(c[r] + bias, 0.f);  // pad -> relu(0)=0
    }
  }
  __syncthreads();

  // ---- Layer 2: [16 x 100] @ w2[100 x 50], 4 N-tiles (waves 0-3) -------
  if (wave < 4) {
    const int ncol = wave * 16 + mrow;                 // < 64
    const float* ap = &h1[mrow * H1_STRIDE + koff];
    const float* bp = &w2f[(half * N2_PAD + ncol) * 2];
    v8f c = {};
#pragma unroll 5
    for (int g = 0; g < K2_STEPS; ++g) {
      v2f a = *(const v2f*)(ap + g * 4);
      v2f b = *(const v2f*)(bp + g * (2 * N2_PAD * 2));
      c = __builtin_amdgcn_wmma_f32_16x16x4_f32(false, a, false, b,
                                                (short)0, c, false, false);
    }
    const float bias = b2p[ncol];
#pragma unroll
    for (int r = 0; r < 8; ++r) {
      int m = r + half * 8;
      h2[m * H2_STRIDE + ncol] = fmaxf(c[r] + bias, 0.f);
    }
  }
  __syncthreads();

  // ---- Layer 3: [16 x 50] @ w3[50 x 25], 2 N-tiles (waves 0-1) ---------
  // K padded to 52: h2 cols 50/51 are zero, w3f rows >= 50 packed as zero.
  if (wave < 2) {
    const int ncol = wave * 16 + mrow;                 // < 32
    const float* ap = &h2[mrow * H2_STRIDE + koff];
    const float* bp = &w3f[(half * N3_PAD + ncol) * 2];
    v8f c = {};
#pragma unroll
    for (int g = 0; g < K3_STEPS; ++g) {
      v2f a = *(const v2f*)(ap + g * 4);
      v2f b = *(const v2f*)(bp + g * (2 * N3_PAD * 2));
      c = __builtin_amdgcn_wmma_f32_16x16x4_f32(false, a, false, b,
                                                (short)0, c, false, false);
    }
    const float bias = b3p[ncol];
#pragma unroll
    for (int r = 0; r < 8; ++r) {
      int m = r + half * 8;
      h3[m * H3_STRIDE + ncol] = fmaxf(c[r] + bias, 0.f);
    }
  }
  __syncthreads();

  // ---- Layer 4: [16 x 25] @ w4[25 x 1] + b4 -> out[base + m] -----------
  if (tid < 16) {
    float acc = b4[0];
#pragma unroll
    for (int j = 0; j < N3; ++j)
      acc += h3[tid * H3_STRIDE + j] * w4[j];
    out[base + tid] = acc;
  }
}

// --------------------------------------------------------------------
extern "C" void kernel_launch(void* const* d_in, const int* in_sizes, int n_in,
                              void* d_out, int out_size, void* d_ws, size_t ws_size,
                              hipStream_t stream) {
  const int*   x     = (const int*)  d_in[0];
  const float* w_emb = (const float*)d_in[1];
  const float* w1    = (const float*)d_in[2];
  const float* b1    = (const float*)d_in[3];
  const float* w2    = (const float*)d_in[4];
  const float* b2    = (const float*)d_in[5];
  const float* w3    = (const float*)d_in[6];
  const float* b3    = (const float*)d_in[7];
  const float* w4    = (const float*)d_in[8];
  const float* b4    = (const float*)d_in[9];

  float* ws  = (float*)d_ws;
  float* S   = ws + S_OFF;
  float* w1f = ws + W1F_OFF;
  float* w2f = ws + W2F_OFF;
  float* w3f = ws + W3F_OFF;
  float* b1p = ws + B1P_OFF;
  float* b2p = ws + B2P_OFF;
  float* b3p = ws + B3P_OFF;

  build_boxsum<<<(TOTAL * EMB + 255) / 256, 256, 0, stream>>>(w_emb, S);
  pack_bfrag<<<(W1F_ELEMS + 255) / 256, 256, 0, stream>>>(w1, w1f, K1, N1, N1_PAD, K1_STEPS);
  pack_bfrag<<<(W2F_ELEMS + 255) / 256, 256, 0, stream>>>(w2, w2f, N1, N2, N2_PAD, K2_STEPS);
  pack_bfrag<<<(W3F_ELEMS + 255) / 256, 256, 0, stream>>>(w3, w3f, N2, N3, N3_PAD, K3_STEPS);
  pack_bias<<<1, N1_PAD, 0, stream>>>(b1, b1p, N1, N1_PAD);
  pack_bias<<<1, N2_PAD, 0, stream>>>(b2, b2p, N2, N2_PAD);
  pack_bias<<<1, N3_PAD, 0, stream>>>(b3, b3p, N3, N3_PAD);

  mlp_fused<<<BATCH / 16, 256, 0, stream>>>(x, S, w1f, b1p, w2f, b2p, w3f, b3p,
                                            w4, b4, (float*)d_out);
}